// GNN_3908420240198
// MI455X (gfx1250) — compile-verified
//
#include <hip/hip_runtime.h>
#include <cstddef>

#define NA_N 5000
#define NC_N 40000
#define E_N  400000

typedef __attribute__((ext_vector_type(2))) float v2f;
typedef __attribute__((ext_vector_type(8))) float v8f;

// ---------- WMMA f32 GEMM: Y[N,NO] = act(X[N,K] @ W[K,NO] + bias) ----------
// One wave (32 threads) per 16x16 output tile. K multiple of 4, NO multiple of 16.
// A 16x4 layout (ISA 7.12.2): lanes 0-15 hold K={k0,k0+1}, lanes 16-31 K={k0+2,k0+3}.
// B 4x16 mirrored over N; C/D: vgpr r = row (r + hi*8), col = lane&15.
__global__ void wmma_linear_kernel(const float* __restrict__ X,
                                   const float* __restrict__ W,
                                   const float* __restrict__ bias,
                                   float* __restrict__ Y,
                                   int N, int K, int NO, int act) {
  int lane = threadIdx.x & 31;
  int m16  = lane & 15;
  int hi   = (lane >= 16) ? 1 : 0;
  int row0 = blockIdx.x * 16;
  int col0 = blockIdx.y * 16;
  int col  = col0 + m16;

  int arow = row0 + m16;
  if (arow >= N) arow = N - 1;           // clamp; stores are guarded
  const float* xp = X + (size_t)arow * K;

  v8f acc = {};
  for (int k0 = 0; k0 < K; k0 += 4) {
    int ka = k0 + (hi ? 2 : 0);
    v2f a, b;
    a.x = xp[ka];
    a.y = xp[ka + 1];
    b.x = W[(size_t)ka * NO + col];
    b.y = W[(size_t)(ka + 1) * NO + col];
    acc = __builtin_amdgcn_wmma_f32_16x16x4_f32(
        /*neg_a=*/false, a, /*neg_b=*/false, b,
        /*c_mod=*/(short)0, acc, /*reuse_a=*/false, /*reuse_b=*/false);
  }

  float bv = bias[col];
#pragma unroll
  for (int r = 0; r < 8; ++r) {
    int row = row0 + r + (hi ? 8 : 0);
    if (row < N) {
      float v = acc[r] + bv;
      if (act == 1) v = v > 0.f ? v : 0.f;   // relu
      Y[(size_t)row * NO + col] = v;
    }
  }
}

// ---------- CSR build: degree count ----------
__global__ void deg_count_kernel(const int* __restrict__ dst,
                                 int* __restrict__ deg, int nE) {
  int e = blockIdx.x * blockDim.x + threadIdx.x;
  if (e >= nE) return;
  atomicAdd(&deg[dst[e]], 1);
}

// ---------- CSR build: single-block exclusive scan (n <= 40960) ----------
__global__ void scan_kernel(const int* __restrict__ deg,
                            int* __restrict__ offs, int n) {
  __shared__ int part[1024];
  int tid = threadIdx.x;
  int chunk = (n + 1023) / 1024;
  int start = tid * chunk;
  int sum = 0;
  for (int i = 0; i < chunk; ++i)
    if (start + i < n) sum += deg[start + i];
  part[tid] = sum;
  __syncthreads();
  for (int off = 1; off < 1024; off <<= 1) {   // inclusive Hillis-Steele
    int v = (tid >= off) ? part[tid - off] : 0;
    __syncthreads();
    part[tid] += v;
    __syncthreads();
  }
  int run = (tid == 0) ? 0 : part[tid - 1];
  for (int i = 0; i < chunk; ++i) {
    if (start + i < n) {
      offs[start + i] = run;
      run += deg[start + i];
    }
  }
  if (tid == 1023) offs[n] = part[1023];
}

// ---------- CSR build: bucket fill ----------
__global__ void csr_fill_kernel(const int* __restrict__ dst,
                                const int* __restrict__ offs,
                                int* __restrict__ cur,
                                int* __restrict__ eid, int nE) {
  int e = blockIdx.x * blockDim.x + threadIdx.x;
  if (e >= nE) return;
  int d = dst[e];
  int pos = offs[d] + atomicAdd(&cur[d], 1);
  eid[pos] = e;
}

// ---------- GATv2 aggregation: one wave per destination node ----------
// lane l owns channels 4l..4l+3 (head = l/16). Atomic-free segment softmax.
// Fuses conv bias + ELU into the final store.
__global__ void gatv2_dst_kernel(const float* __restrict__ xl,   // [Ns,128]
                                 const float* __restrict__ xr,   // [Nd,128]
                                 const int* __restrict__ offs,   // [Nd+1]
                                 const int* __restrict__ eid,    // [E]
                                 const int* __restrict__ src,    // [E]
                                 const float* __restrict__ att,  // [128]
                                 const float* __restrict__ We,   // [128] or null
                                 const float* __restrict__ eattr,// [E] or null
                                 const float* __restrict__ bias, // [128]
                                 float* __restrict__ out,        // [Nd,128]
                                 float* __restrict__ sc,         // [E*2]
                                 int Nd) {
  int wave = (blockIdx.x * blockDim.x + threadIdx.x) >> 5;
  int lane = threadIdx.x & 31;
  if (wave >= Nd) return;
  int d = wave;
  int base = offs[d];
  int deg  = offs[d + 1] - base;
  int h = lane >> 4;

  float4 xr4 = ((const float4*)(xr + (size_t)d * 128))[lane];
  float4 at4 = ((const float4*)att)[lane];
  float4 we4 = make_float4(0.f, 0.f, 0.f, 0.f);
  if (We) we4 = ((const float4*)We)[lane];

  // ---- pass 1: per-edge scores -> sc, running per-head max in registers ----
  float mx0 = -3.4e38f, mx1 = -3.4e38f;
  for (int i = 0; i < deg; ++i) {
    int e = eid[base + i];
    int s = src[e];
    float ea = We ? eattr[e] : 0.f;
    float4 xl4 = ((const float4*)(xl + (size_t)s * 128))[lane];
    float m0 = xl4.x + xr4.x + ea * we4.x; m0 = m0 > 0.f ? m0 : 0.2f * m0;
    float m1 = xl4.y + xr4.y + ea * we4.y; m1 = m1 > 0.f ? m1 : 0.2f * m1;
    float m2 = xl4.z + xr4.z + ea * we4.z; m2 = m2 > 0.f ? m2 : 0.2f * m2;
    float m3 = xl4.w + xr4.w + ea * we4.w; m3 = m3 > 0.f ? m3 : 0.2f * m3;
    float p = m0 * at4.x + m1 * at4.y + m2 * at4.z + m3 * at4.w;
    p += __shfl_xor(p, 1, 32);
    p += __shfl_xor(p, 2, 32);
    p += __shfl_xor(p, 4, 32);
    p += __shfl_xor(p, 8, 32);                 // reduce within each 16-lane head
    float s0 = __shfl(p, 0, 32);
    float s1 = __shfl(p, 16, 32);
    if (lane == 0)  sc[(size_t)(base + i) * 2 + 0] = s0;
    if (lane == 16) sc[(size_t)(base + i) * 2 + 1] = s1;
    mx0 = fmaxf(mx0, s0);
    mx1 = fmaxf(mx1, s1);
  }

  // ---- pass 2: denominators (register reduction, no atomics) ----
  float dn0 = 0.f, dn1 = 0.f;
  for (int i = lane; i < deg; i += 32) {
    dn0 += __expf(sc[(size_t)(base + i) * 2 + 0] - mx0);
    dn1 += __expf(sc[(size_t)(base + i) * 2 + 1] - mx1);
  }
#pragma unroll
  for (int m = 1; m < 32; m <<= 1) {
    dn0 += __shfl_xor(dn0, m, 32);
    dn1 += __shfl_xor(dn1, m, 32);
  }
  float mxh = (h == 0) ? mx0 : mx1;
  float inv = 1.f / (((h == 0) ? dn0 : dn1) + 1e-16f);

  // ---- pass 3: out[d] = elu(sum_e xl[src]*alpha + bias) ----
  float4 acc = make_float4(0.f, 0.f, 0.f, 0.f);
  for (int i = 0; i < deg; ++i) {
    int e = eid[base + i];
    int s = src[e];
    float a = __expf(sc[(size_t)(base + i) * 2 + h] - mxh) * inv;
    float4 xl4 = ((const float4*)(xl + (size_t)s * 128))[lane];
    acc.x += xl4.x * a;
    acc.y += xl4.y * a;
    acc.z += xl4.z * a;
    acc.w += xl4.w * a;
  }
  float4 bv = ((const float4*)bias)[lane];
  float4 o;
  o.x = acc.x + bv.x; o.x = o.x > 0.f ? o.x : __expf(o.x) - 1.f;
  o.y = acc.y + bv.y; o.y = o.y > 0.f ? o.y : __expf(o.y) - 1.f;
  o.z = acc.z + bv.z; o.z = o.z > 0.f ? o.z : __expf(o.z) - 1.f;
  o.w = acc.w + bv.w; o.w = o.w > 0.f ? o.w : __expf(o.w) - 1.f;
  ((float4*)(out + (size_t)d * 128))[lane] = o;
}

// ---------- tiny head output: Y[N,NO] = X[N,64] @ W[64,NO] + b ----------
__global__ void head_out_kernel(const float* __restrict__ X,
                                const float* __restrict__ W,
                                const float* __restrict__ b,
                                float* __restrict__ Y, int N, int NO) {
  int idx = blockIdx.x * blockDim.x + threadIdx.x;
  if (idx >= N * NO) return;
  int i = idx / NO, j = idx % NO;
  float s = b[j];
#pragma unroll 8
  for (int k = 0; k < 64; ++k) s += X[(size_t)i * 64 + k] * W[(size_t)k * NO + j];
  Y[idx] = s;
}

extern "C" void kernel_launch(void* const* d_in, const int* in_sizes, int n_in,
                              void* d_out, int out_size, void* d_ws, size_t ws_size,
                              hipStream_t stream) {
  (void)in_sizes; (void)n_in; (void)out_size; (void)ws_size;

  const float* x_ap    = (const float*)d_in[0];
  const float* x_cl    = (const float*)d_in[1];
  const int*   a2c_src = (const int*)d_in[2];
  const int*   a2c_dst = (const int*)d_in[3];
  const int*   c2a_src = (const int*)d_in[4];
  const int*   c2a_dst = (const int*)d_in[5];
  const float* eattr   = (const float*)d_in[6];
  const float* enc_ap_W = (const float*)d_in[7];
  const float* enc_ap_b = (const float*)d_in[8];
  const float* enc_cl_W = (const float*)d_in[9];
  const float* enc_cl_b = (const float*)d_in[10];
  // relation weight blocks: c1_a2c @11 (7 tensors), c1_c2a @18 (6), c2_a2c @24 (7, dead), c2_c2a @31 (6)
  const float* ch_W1 = (const float*)d_in[37];
  const float* ch_b1 = (const float*)d_in[38];
  const float* ch_W2 = (const float*)d_in[39];
  const float* ch_b2 = (const float*)d_in[40];
  const float* pw_W1 = (const float*)d_in[41];
  const float* pw_b1 = (const float*)d_in[42];
  const float* pw_W2 = (const float*)d_in[43];
  const float* pw_b2 = (const float*)d_in[44];

  // ---- workspace layout (float units, 1KB-aligned blocks) ----
  float* ws = (float*)d_ws;
  size_t off = 0;
  auto A = [&](size_t n) { size_t o = off; off += (n + 255) & ~(size_t)255; return o; };
  float* H_AP = ws + A((size_t)NA_N * 64);
  float* H_CL = ws + A((size_t)NC_N * 64);
  float* CL1  = ws + A((size_t)NC_N * 128);
  float* AP1  = ws + A((size_t)NA_N * 128);
  float* AP2  = ws + A((size_t)NA_N * 128);
  float* XL   = ws + A((size_t)NC_N * 128);
  float* XR   = ws + A((size_t)NC_N * 128);
  float* SC   = ws + A((size_t)E_N * 2);
  int* OFFS_C = (int*)(ws + A(NC_N + 1));
  int* CUR_C  = (int*)(ws + A(NC_N));
  int* EID_AC = (int*)(ws + A(E_N));
  int* OFFS_A = (int*)(ws + A(NA_N + 1));
  int* CUR_A  = (int*)(ws + A(NA_N));
  int* EID_CA = (int*)(ws + A(E_N));
  float* HID  = H_AP;  // reuse: h_ap dead after conv1

  dim3 b32(32);
  auto lin = [&](const float* X, int N, int K, const float* W, const float* bias,
                 int NO, int act, float* Y) {
    dim3 g((N + 15) / 16, NO / 16);
    wmma_linear_kernel<<<g, b32, 0, stream>>>(X, W, bias, Y, N, K, NO, act);
  };

  const int TB = 256;
  const int EG = (E_N + TB - 1) / TB;

  // ---- build CSR-by-destination for both relations (reused across convs) ----
  auto build_csr = [&](const int* dstArr, int Nd, int* offs, int* cur, int* eid) {
    hipMemsetAsync(cur, 0, (size_t)Nd * sizeof(int), stream);
    deg_count_kernel<<<EG, TB, 0, stream>>>(dstArr, cur, E_N);
    scan_kernel<<<1, 1024, 0, stream>>>(cur, offs, Nd);
    hipMemsetAsync(cur, 0, (size_t)Nd * sizeof(int), stream);
    csr_fill_kernel<<<EG, TB, 0, stream>>>(dstArr, offs, cur, eid, E_N);
  };
  build_csr(a2c_dst, NC_N, OFFS_C, CUR_C, EID_AC);
  build_csr(c2a_dst, NA_N, OFFS_A, CUR_A, EID_CA);

  // ---- one GATv2 conv (dst-centric, atomic-free) ----
  auto conv = [&](const float* xs, int Ns, const float* xd, int Nd, int K,
                  const int* offs, const int* eid, const int* es,
                  int wi, bool has_edge, float* outp) {
    const float* Wl   = (const float*)d_in[wi + 0];
    const float* bl   = (const float*)d_in[wi + 1];
    const float* Wr   = (const float*)d_in[wi + 2];
    const float* br   = (const float*)d_in[wi + 3];
    const float* att  = (const float*)d_in[wi + 4];
    const float* bias = (const float*)d_in[wi + 5];
    const float* We   = has_edge ? (const float*)d_in[wi + 6] : nullptr;
    lin(xs, Ns, K, Wl, bl, 128, 0, XL);     // xl = x_src @ Wl + bl
    lin(xd, Nd, K, Wr, br, 128, 0, XR);     // xr = x_dst @ Wr + br
    int waves_per_block = TB / 32;
    int grid = (Nd + waves_per_block - 1) / waves_per_block;
    gatv2_dst_kernel<<<grid, TB, 0, stream>>>(XL, XR, offs, eid, es, att, We,
                                              eattr, bias, outp, SC, Nd);
  };

  // encoders (WMMA, K=8 / K=4)
  lin(x_ap, NA_N, 8, enc_ap_W, enc_ap_b, 64, 0, H_AP);
  lin(x_cl, NC_N, 4, enc_cl_W, enc_cl_b, 64, 0, H_CL);

  // conv1: ap->client and client->ap ; conv2: only client->ap (cl2 is dead)
  conv(H_AP, NA_N, H_CL, NC_N, 64,  OFFS_C, EID_AC, a2c_src, 11, true,  CL1);
  conv(H_CL, NC_N, H_AP, NA_N, 64,  OFFS_A, EID_CA, c2a_src, 18, false, AP1);
  conv(CL1,  NC_N, AP1,  NA_N, 128, OFFS_A, EID_CA, c2a_src, 31, false, AP2);

  // output heads (layer1 via WMMA + relu, layer2 scalar into d_out)
  float* outf = (float*)d_out;
  lin(AP2, NA_N, 128, ch_W1, ch_b1, 64, 1, HID);
  head_out_kernel<<<(NA_N * 3 + TB - 1) / TB, TB, 0, stream>>>(HID, ch_W2, ch_b2, outf, NA_N, 3);
  lin(AP2, NA_N, 128, pw_W1, pw_b1, 64, 1, HID);
  head_out_kernel<<<(NA_N * 4 + TB - 1) / TB, TB, 0, stream>>>(HID, pw_W2, pw_b2, outf + NA_N * 3, NA_N, 4);
}